// PointsMasksMatcher_35570919145557
// MI455X (gfx1250) — compile-verified
//
#include <hip/hip_runtime.h>
#include <hip/hip_bf16.h>
#include <stdint.h>

// Problem sizes (fixed by the reference)
#define BATCH 8
#define NP    4096   // P predicted points
#define NG    1024   // N ground-truth entries
#define HH    1024
#define WW    1024

#define TPB    512   // threads per block
#define WAVES  16    // wave32 waves per block
#define TILES  16    // 16x16 WMMA tiles per wave (256 columns per wave)

typedef float v2f __attribute__((ext_vector_type(2)));
typedef float v8f __attribute__((ext_vector_type(8)));

static __device__ __forceinline__ unsigned long long umin64(unsigned long long a,
                                                            unsigned long long b) {
  return a < b ? a : b;
}

// One workgroup per batch. 512 threads = 16 wave32 waves (4 waves/SIMD ->
// ~256 VGPRs/wave, so the 128 accumulator + 32 B-operand VGPRs stay resident).
// Each wave owns 256 point-columns (16 WMMA tiles of 16 cols).
// The 16x4096 d^2 chunk lives entirely in WMMA accumulator VGPRs.
__global__ __launch_bounds__(TPB, 1)
void PointsMasksMatcher_kernel(const float* __restrict__ predPts,   // [B,P,2]
                               const float* __restrict__ gtPts,     // [B,N,2]
                               const int*   __restrict__ gtKeys,    // [B,N]
                               const int*   __restrict__ masks,     // [B,H,W]
                               float*       __restrict__ out)       // src|tgt|cost
{
  const int b    = blockIdx.x;
  const int tid  = threadIdx.x;
  const int wave = tid >> 5;
  const int lane = tid & 31;
  const int half = lane >> 4;   // 0: lanes 0-15, 1: lanes 16-31
  const int lq   = lane & 15;

  __shared__ float sVx[NG], sVy[NG], sVn[NG];
  __shared__ int   sKeys[NG];
  __shared__ unsigned sPresent[33];                 // bitset over label ids 0..1056
  __shared__ unsigned long long redIn[WAVES], redBg[WAVES];
  __shared__ int redCnt[WAVES];
  __shared__ unsigned long long finIn, finBg;
  __shared__ int finCnt;

  // ---- cooperative loads ----
  for (int i = tid; i < NG; i += TPB) {
    float vx = gtPts[(b * NG + i) * 2 + 0];
    float vy = gtPts[(b * NG + i) * 2 + 1];
    sVx[i] = vx; sVy[i] = vy; sVn[i] = vx * vx + vy * vy;
    sKeys[i] = gtKeys[b * NG + i];
  }
  if (tid < 33) sPresent[tid] = 0u;
  __syncthreads();
  for (int i = tid; i < NG; i += TPB) {
    int k1 = sKeys[i] + 1;
    atomicOr(&sPresent[k1 >> 5], 1u << (k1 & 31));
  }

  // ---- per-lane columns: labels (packed 2x16b) + WMMA B operand prep ----
  unsigned labPk[TILES / 2];         // 16 labels, 16 bits each
  v2f      Bop[TILES];               // per-tile B operand (K pair for this lane-half)
  unsigned availBits = 0, bgBits = 0;
  const int colLane = wave * (TILES * 16) + lq;   // col of tile t = colLane + 16*t
#pragma unroll
  for (int t = 0; t < TILES; ++t) {
    int c = colLane + t * 16;                      // both lane-halves load same cols
    float ux = predPts[(b * NP + c) * 2 + 0];
    float uy = predPts[(b * NP + c) * 2 + 1];
    int xi = (int)rintf(ux * (float)HH); xi = xi < 0 ? 0 : (xi > HH - 1 ? HH - 1 : xi);
    int yi = (int)rintf(uy * (float)WW); yi = yi < 0 ? 0 : (yi > WW - 1 ? WW - 1 : yi);
    unsigned L = (unsigned)masks[(size_t)b * HH * WW + (size_t)xi * WW + yi];
    if ((t & 1) == 0) labPk[t >> 1] = L;
    else              labPk[t >> 1] |= L << 16;
    // B column = (-2Ux, -2Uy, 1, |U|^2)^T ; K0/K1 in lanes 0-15, K2/K3 in lanes 16-31
    Bop[t].x = half ? 1.0f : -2.0f * ux;
    Bop[t].y = half ? (ux * ux + uy * uy) : -2.0f * uy;
  }
  __syncthreads();  // presence table complete
#pragma unroll
  for (int t = 0; t < TILES; ++t) {
    unsigned L = (labPk[t >> 1] >> ((t & 1) * 16)) & 0xFFFFu;
    bool inAny = (sPresent[L >> 5] >> (L & 31)) & 1u;
    if (inAny) availBits |= 1u << t; else bgBits |= 1u << t;
  }

  float* oSrc  = out;
  float* oTgt  = out + BATCH * NG;
  float* oCost = out + 2 * BATCH * NG;
  float  cost  = 0.0f;
  const unsigned long long SENT = ~0ull;

  // ---- greedy scan, 16 GT rows per WMMA chunk ----
  for (int j0 = 0; j0 < NG; j0 += 16) {
    // A row = (Vx, Vy, |V|^2, 1): K0/K1 lanes 0-15, K2/K3 lanes 16-31
    v2f A;
    A.x = half ? sVn[j0 + lq] : sVx[j0 + lq];
    A.y = half ? 1.0f         : sVy[j0 + lq];

    v8f dsq[TILES];
#pragma unroll
    for (int t = 0; t < TILES; ++t) {
      v8f cz = {};
      // d^2 tile = A(16x4) x B(4x16) + 0  -> v_wmma_f32_16x16x4_f32
      dsq[t] = __builtin_amdgcn_wmma_f32_16x16x4_f32(
          false, A, false, Bop[t], (short)0, cz, false, false);
    }

    for (int jj = 0; jj < 16; ++jj) {
      const int j    = j0 + jj;
      const int key1 = sKeys[j] + 1;
      // Accumulator layout: VGPR r holds row r (lanes 0-15) and row r+8 (lanes 16-31)
      const bool active = (half == (jj >> 3));
      const int  rowr   = jj & 7;

      unsigned long long mIn = SENT, mBg = SENT;
      int mCnt = 0;
      if (active) {
        // per-lane candidate / bg masks for this row
        unsigned candMask = 0;
#pragma unroll
        for (int t = 0; t < TILES; ++t) {
          unsigned L = (labPk[t >> 1] >> ((t & 1) * 16)) & 0xFFFFu;
          if (L == (unsigned)key1) candMask |= 1u << t;
        }
        candMask &= availBits;
        mCnt = __popc(candMask);
        // 4 parallel min-chains to bound temp pressure (dsq+Bop stay resident)
        unsigned long long aIn[4] = {SENT, SENT, SENT, SENT};
        unsigned long long aBg[4] = {SENT, SENT, SENT, SENT};
#pragma unroll
        for (int t = 0; t < TILES; ++t) {
          float v = fmaxf(dsq[t][rowr], 0.0f);  // keep u64 packing order-preserving
          unsigned long long pack =
              ((unsigned long long)__float_as_uint(v) << 32) |
              (unsigned)(colLane + t * 16);
          if ((candMask >> t) & 1u) aIn[t & 3] = umin64(aIn[t & 3], pack);
          if ((bgBits   >> t) & 1u) aBg[t & 3] = umin64(aBg[t & 3], pack);
        }
        mIn = umin64(umin64(aIn[0], aIn[1]), umin64(aIn[2], aIn[3]));
        mBg = umin64(umin64(aBg[0], aBg[1]), umin64(aBg[2], aBg[3]));
      }
      // wave32 reduction
#pragma unroll
      for (int off = 16; off > 0; off >>= 1) {
        mIn  = umin64(mIn, __shfl_xor(mIn, off, 32));
        mBg  = umin64(mBg, __shfl_xor(mBg, off, 32));
        mCnt += __shfl_xor(mCnt, off, 32);
      }
      if (lane == 0) { redIn[wave] = mIn; redBg[wave] = mBg; redCnt[wave] = mCnt; }
      __syncthreads();
      if (wave == 0) {
        unsigned long long rIn = (lane < WAVES) ? redIn[lane] : SENT;
        unsigned long long rBg = (lane < WAVES) ? redBg[lane] : SENT;
        int rC = (lane < WAVES) ? redCnt[lane] : 0;
#pragma unroll
        for (int off = 16; off > 0; off >>= 1) {
          rIn = umin64(rIn, __shfl_xor(rIn, off, 32));
          rBg = umin64(rBg, __shfl_xor(rBg, off, 32));
          rC  += __shfl_xor(rC, off, 32);
        }
        if (lane == 0) { finIn = rIn; finBg = rBg; finCnt = rC; }
      }
      __syncthreads();

      const unsigned long long fIn = finIn, fBg = finBg;
      const int  fC    = finCnt;
      const bool useIn = fC > 0;
      const bool hasBg = fBg != SENT;
      const int  u = useIn ? (int)(fIn & 0xFFFFFFFFu)
                           : (hasBg ? (int)(fBg & 0xFFFFFFFFu) : -1);
      if (tid == 0) {
        float c = useIn ? sqrtf(__uint_as_float((unsigned)(fIn >> 32)))
                        : (hasBg ? sqrtf(__uint_as_float((unsigned)(fBg >> 32))) : 0.0f);
        cost += c;
        oSrc[b * NG + j] = (float)u;
        oTgt[b * NG + j] = (u >= 0) ? (float)j : -1.0f;
      }
      // state update (both lane-halves keep identical copies of column state)
      if (useIn) {
#pragma unroll
        for (int t = 0; t < TILES; ++t) {
          unsigned L = (labPk[t >> 1] >> ((t & 1) * 16)) & 0xFFFFu;
          bool cand  = (L == (unsigned)key1) && ((availBits >> t) & 1u);
          int  c     = colLane + t * 16;
          if (c == u)                    availBits &= ~(1u << t);
          if (fC > 1 && cand && c != u)  bgBits    |=  (1u << t);
        }
      } else if (hasBg) {
#pragma unroll
        for (int t = 0; t < TILES; ++t)
          if ((colLane + t * 16) == u) bgBits &= ~(1u << t);
      }
    }
  }
  if (tid == 0) oCost[b] = cost;
}

extern "C" void kernel_launch(void* const* d_in, const int* in_sizes, int n_in,
                              void* d_out, int out_size, void* d_ws, size_t ws_size,
                              hipStream_t stream) {
  const float* predPts = (const float*)d_in[0];   // [8,4096,2] f32
  const float* gtPts   = (const float*)d_in[1];   // [8,1024,2] f32
  const int*   gtKeys  = (const int*)  d_in[2];   // [8,1024]   i32
  const int*   masksP  = (const int*)  d_in[3];   // [8,1024,1024] i32
  float* out = (float*)d_out;                     // src(8192)|tgt(8192)|cost(8)

  PointsMasksMatcher_kernel<<<dim3(BATCH), dim3(TPB), 0, stream>>>(
      predPts, gtPts, gtKeys, masksP, out);
}